// AttentionNCF_68315749810659
// MI455X (gfx1250) — compile-verified
//
#include <hip/hip_runtime.h>
#include <hip/hip_bf16.h>

typedef float v2f __attribute__((ext_vector_type(2)));
typedef float v8f __attribute__((ext_vector_type(8)));

// ---- CDNA5 async global->LDS support (guarded; falls back to sync staging) --
#if __has_builtin(__builtin_amdgcn_global_load_async_to_lds_b32)
#define HAVE_ASYNC_LDS 1
#else
#define HAVE_ASYNC_LDS 0
#endif

// builtin prototype (from hipcc diagnostic): (__device__ int*, __shared__ int*, Ii, Ii)
typedef __attribute__((address_space(1))) int g_i32;
typedef __attribute__((address_space(3))) int l_i32;

__device__ __forceinline__ g_i32* glb_i32(const float* p) {
    return (g_i32*)(unsigned long long)p;           // strips const via int round-trip
}
__device__ __forceinline__ l_i32* lds_i32(float* p) {
    // generic LDS addresses carry the wave-relative LDS offset in addr[31:0]
    return (l_i32*)(unsigned)(unsigned long long)p;
}

__device__ __forceinline__ void wait_async_le12() {
#if __has_builtin(__builtin_amdgcn_s_wait_asynccnt)
    __builtin_amdgcn_s_wait_asynccnt(12);
#else
    asm volatile("s_wait_asynccnt 0xc" ::: "memory");
#endif
}
__device__ __forceinline__ void wait_async_0() {
#if __has_builtin(__builtin_amdgcn_s_wait_asynccnt)
    __builtin_amdgcn_s_wait_asynccnt(0);
#else
    asm volatile("s_wait_asynccnt 0x0" ::: "memory");
#endif
}

// ---------------------------------------------------------------------------
// out[row] = dot(X[row, :ncols], w) + (addend ? addend[0] : 0)
// one wave (32 lanes) per row; blockDim = 256 -> 8 rows per block
// ---------------------------------------------------------------------------
__global__ __launch_bounds__(256) void rowdot_kernel(
    const float* __restrict__ X, const float* __restrict__ w,
    const float* __restrict__ addend, float* __restrict__ out, int ncols)
{
    const int wave = threadIdx.x >> 5;
    const int lane = threadIdx.x & 31;
    const int row  = blockIdx.x * 8 + wave;
    const float* xr = X + (size_t)row * ncols;
    float s = 0.0f;
    for (int c = lane; c < ncols; c += 32) s += xr[c] * w[c];
    #pragma unroll
    for (int off = 16; off > 0; off >>= 1) s += __shfl_xor(s, off, 32);
    if (lane == 0) out[row] = s + (addend ? addend[0] : 0.0f);
}

// ---------------------------------------------------------------------------
// Masked softmax over I rated items, fused with the elementwise multiply:
//   W[b,i] = softmax_i( cs[b] + rs[i] + att_b  where um!=0 else -inf ) * um[b,i]
// One 256-thread block per row b.
// ---------------------------------------------------------------------------
__global__ __launch_bounds__(256) void attn_weights_kernel(
    const float* __restrict__ um, const float* __restrict__ cs,
    const float* __restrict__ rs, const float* __restrict__ attb,
    float* __restrict__ W, int I)
{
    const int b   = blockIdx.x;
    const int tid = threadIdx.x;
    const float* umrow = um + (size_t)b * I;
    const float cb = cs[b] + attb[0];
    __shared__ float red[256];

    float mx = -INFINITY;
    for (int i = tid; i < I; i += 256) {
        float u = umrow[i];
        if (u != 0.0f) mx = fmaxf(mx, cb + rs[i]);
    }
    red[tid] = mx; __syncthreads();
    for (int s = 128; s > 0; s >>= 1) {
        if (tid < s) red[tid] = fmaxf(red[tid], red[tid + s]);
        __syncthreads();
    }
    mx = red[0];
    __syncthreads();

    float sum = 0.0f;
    if (mx > -INFINITY) {
        for (int i = tid; i < I; i += 256) {
            float u = umrow[i];
            if (u != 0.0f) sum += __expf(cb + rs[i] - mx);
        }
    }
    red[tid] = sum; __syncthreads();
    for (int s = 128; s > 0; s >>= 1) {
        if (tid < s) red[tid] += red[tid + s];
        __syncthreads();
    }
    sum = red[0];
    const float inv = (sum > 0.0f) ? 1.0f / sum : 0.0f;

    for (int i = tid; i < I; i += 256) {
        float u = umrow[i];
        float wv = 0.0f;
        if (u != 0.0f) wv = __expf(cb + rs[i] - mx) * inv * u;
        W[(size_t)b * I + i] = wv;
    }
}

// ---------------------------------------------------------------------------
// fp32 WMMA GEMM:  C[M,N] = act( A[M,K] @ B[K,N] + bias )
// Block tile 128x64, 256 threads = 8 waves; each wave owns a 16x64 strip
// (4 accumulators of 16x16, v_wmma_f32_16x16x4_f32, K chunked by 16).
//
// LDS holds fragment-native layouts so every WMMA operand is ONE contiguous
// 8-byte-aligned ds_load_b64 (no VGPR shuffles):
//   A: As[r][c] at float index r*20 + c      (row stride 20 -> conflict-free)
//   B: elem(k,n) at (k>>1)*160 + n*2 + (k&1) (k-pair groups, stride 160 ->
//      the two lane-halves hit disjoint bank sets, offset 160 % 64 == 32)
// Chunks are staged with GLOBAL_LOAD_ASYNC_TO_LDS_B32 (ASYNCcnt, per-lane LDS
// destination does the swizzle for free), double-buffered so the memory pipe
// overlaps the matrix pipe.
// Requires M%128==0, N%64==0, K%16==0 (true for every call below) -> EXEC is
// all-ones at every WMMA as the ISA requires.
// ---------------------------------------------------------------------------
#define TM 128
#define TN 64
#define TK 16
#define ASTR 20                // A row stride in LDS (floats)
#define BGRP 160               // B k-pair group stride in LDS (floats)
#define ABUF (TM * ASTR)       // 2560 floats
#define BBUF ((TK / 2) * BGRP) // 1280 floats
#define CHUNK (ABUF + BBUF)    // 3840 floats per buffer

__device__ __forceinline__ void stage_chunk(
    const float* __restrict__ A, int lda,
    const float* __restrict__ B, int ldb,
    int m0, int n0, int k0, int tid, float* sbuf)
{
#if HAVE_ASYNC_LDS
    #pragma unroll
    for (int it = 0; it < (TM * TK) / 256; ++it) {     // 8 issues
        int idx = tid + it * 256;
        int r = idx >> 4, c = idx & 15;
        __builtin_amdgcn_global_load_async_to_lds_b32(
            glb_i32(A + (size_t)(m0 + r) * lda + (k0 + c)),
            lds_i32(sbuf + r * ASTR + c), 0, 0);
    }
    float* sB = sbuf + ABUF;
    #pragma unroll
    for (int it = 0; it < (TK * TN) / 256; ++it) {     // 4 issues
        int idx = tid + it * 256;
        int k = idx >> 6, n = idx & 63;
        __builtin_amdgcn_global_load_async_to_lds_b32(
            glb_i32(B + (size_t)(k0 + k) * ldb + (n0 + n)),
            lds_i32(sB + (k >> 1) * BGRP + n * 2 + (k & 1)), 0, 0);
    }
#else
    #pragma unroll
    for (int it = 0; it < (TM * TK) / 256; ++it) {
        int idx = tid + it * 256;
        int r = idx >> 4, c = idx & 15;
        sbuf[r * ASTR + c] = A[(size_t)(m0 + r) * lda + (k0 + c)];
    }
    float* sB = sbuf + ABUF;
    #pragma unroll
    for (int it = 0; it < (TK * TN) / 256; ++it) {
        int idx = tid + it * 256;
        int k = idx >> 6, n = idx & 63;
        sB[(k >> 1) * BGRP + n * 2 + (k & 1)] = B[(size_t)(k0 + k) * ldb + (n0 + n)];
    }
#endif
}

__global__ __launch_bounds__(256) void gemm_wmma_f32(
    const float* __restrict__ A, int lda,
    const float* __restrict__ B, int ldb,
    const float* __restrict__ bias,
    float* __restrict__ C, int ldc,
    int M, int N, int K, int do_relu)
{
    __shared__ float smem[2 * CHUNK];

    const int tid  = threadIdx.x;
    const int wave = tid >> 5;
    const int lane = tid & 31;
    const int half = lane >> 4;       // 0: lanes 0-15, 1: lanes 16-31
    const int lr   = lane & 15;
    const int m0   = blockIdx.y * TM;
    const int n0   = blockIdx.x * TN;

    v8f acc[4] = {};

    const int nk = K / TK;
    stage_chunk(A, lda, B, ldb, m0, n0, 0, tid, smem);

    for (int kc = 0; kc < nk; ++kc) {
        if (kc + 1 < nk) {
            stage_chunk(A, lda, B, ldb, m0, n0, (kc + 1) * TK, tid,
                        smem + ((kc + 1) & 1) * CHUNK);
            wait_async_le12();      // chunk kc drained; kc+1 may stay in flight
        } else {
            wait_async_0();
        }
        __syncthreads();

        const float* As = smem + (kc & 1) * CHUNK;
        const float* Bs = As + ABUF;

        #pragma unroll
        for (int kk = 0; kk < TK; kk += 4) {
            const int ak = kk + half * 2;
            // A fragment (16x4 f32): lanes 0-15 K={kk,kk+1}, lanes 16-31 K={kk+2,kk+3}
            v2f a = *(const v2f*)(As + (wave * 16 + lr) * ASTR + ak);
            #pragma unroll
            for (int t = 0; t < 4; ++t) {
                // B fragment (4x16): contiguous k-pair for column n = t*16+lr
                v2f bf = *(const v2f*)(Bs + (ak >> 1) * BGRP + (t * 16 + lr) * 2);
                acc[t] = __builtin_amdgcn_wmma_f32_16x16x4_f32(
                    false, a, false, bf, (short)0, acc[t], false, false);
            }
        }
        __syncthreads();
    }

    // store: C/D layout -> VGPR r, lanes 0-15: M=r, lanes 16-31: M=8+r
    const int row_base = m0 + wave * 16 + half * 8;
    #pragma unroll
    for (int t = 0; t < 4; ++t) {
        const int col = t * 16 + lr;
        const float bv = bias ? bias[n0 + col] : 0.0f;
        #pragma unroll
        for (int r = 0; r < 8; ++r) {
            float v = acc[t][r] + bv;
            if (do_relu) v = fmaxf(v, 0.0f);
            C[(size_t)(row_base + r) * ldc + (n0 + col)] = v;
        }
    }
}

// ---------------------------------------------------------------------------
extern "C" void kernel_launch(void* const* d_in, const int* in_sizes, int n_in,
                              void* d_out, int out_size, void* d_ws, size_t ws_size,
                              hipStream_t stream)
{
    const float* cand  = (const float*)d_in[0];   // (2048, 512)
    const float* rated = (const float*)d_in[1];   // (4096, 512)
    const float* um    = (const float*)d_in[2];   // (2048, 4096)
    const float* attw  = (const float*)d_in[3];   // (1024, 1)
    const float* attb  = (const float*)d_in[4];   // (1,)
    const float* iw1   = (const float*)d_in[5];   // (512, 1024)
    const float* ib1   = (const float*)d_in[6];
    const float* iw2   = (const float*)d_in[7];   // (1024, 512)
    const float* ib2   = (const float*)d_in[8];
    const float* uw1   = (const float*)d_in[9];   // (512, 2048)
    const float* ub1   = (const float*)d_in[10];
    const float* uw2   = (const float*)d_in[11];  // (2048, 1024)
    const float* ub2   = (const float*)d_in[12];
    const float* mw1   = (const float*)d_in[13];  // (1536, 1024)
    const float* mb1   = (const float*)d_in[14];
    const float* mw2   = (const float*)d_in[15];  // (1024, 512)
    const float* mb2   = (const float*)d_in[16];
    const float* mw3   = (const float*)d_in[17];  // (512, 256)
    const float* mb3   = (const float*)d_in[18];
    const float* mw4   = (const float*)d_in[19];  // (256, 1)
    const float* mb4   = (const float*)d_in[20];

    constexpr int Bn = 2048, I = 4096, D = 512;
    constexpr int IE = 512, UE = 1024;
    constexpr int D1 = 1024, D2 = 512, D3 = 256;

    // workspace layout (floats); hbuf reuses the W region once W is consumed
    float* ws    = (float*)d_ws;
    float* W     = ws;                              // 8M  floats (2048x4096)
    float* hbuf  = W;                               // reused: h_i1 / h_u1 / x1 / x3
    float* ufeat = ws + (size_t)Bn * I;             // 1M  floats (2048x512)
    float* xcat  = ufeat + (size_t)Bn * D;          // 3M  floats (2048x1536)
    float* x2buf = xcat + (size_t)Bn * (IE + UE);   // 1M  floats (2048x512)
    float* cs    = x2buf + (size_t)Bn * D2;         // 2048
    float* rs    = cs + Bn;                         // 4096

    const dim3 blk(256);
    auto grid = [](int N, int M) { return dim3((unsigned)(N / TN), (unsigned)(M / TM)); };

    // 1) attention scores per candidate / per rated item
    rowdot_kernel<<<Bn / 8, blk, 0, stream>>>(cand,  attw,     nullptr, cs, D);
    rowdot_kernel<<<I  / 8, blk, 0, stream>>>(rated, attw + D, nullptr, rs, D);

    // 2) masked softmax fused with elementwise multiply -> W
    attn_weights_kernel<<<Bn, blk, 0, stream>>>(um, cs, rs, attb, W, I);

    // 3) user_feat = W @ rated_items                      (2048x4096 @ 4096x512)
    gemm_wmma_f32<<<grid(D, Bn), blk, 0, stream>>>(W, I, rated, D, nullptr,
                                                   ufeat, D, Bn, D, I, 0);
    // 4) h_i1 = relu(cand @ iw1 + ib1)                    (K=512, N=1024)
    gemm_wmma_f32<<<grid(2 * IE, Bn), blk, 0, stream>>>(cand, D, iw1, 2 * IE, ib1,
                                                        hbuf, 2 * IE, Bn, 2 * IE, D, 1);
    // 5) item_emb = relu(h_i1 @ iw2 + ib2) -> xcat[:, 0:512]
    gemm_wmma_f32<<<grid(IE, Bn), blk, 0, stream>>>(hbuf, 2 * IE, iw2, IE, ib2,
                                                    xcat, IE + UE, Bn, IE, 2 * IE, 1);
    // 6) h_u1 = relu(ufeat @ uw1 + ub1)                   (K=512, N=2048)
    gemm_wmma_f32<<<grid(2 * UE, Bn), blk, 0, stream>>>(ufeat, D, uw1, 2 * UE, ub1,
                                                        hbuf, 2 * UE, Bn, 2 * UE, D, 1);
    // 7) user_emb = relu(h_u1 @ uw2 + ub2) -> xcat[:, 512:1536]
    gemm_wmma_f32<<<grid(UE, Bn), blk, 0, stream>>>(hbuf, 2 * UE, uw2, UE, ub2,
                                                    xcat + IE, IE + UE, Bn, UE, 2 * UE, 1);
    // 8) x1 = relu(xcat @ mw1 + mb1)                      (K=1536, N=1024)
    gemm_wmma_f32<<<grid(D1, Bn), blk, 0, stream>>>(xcat, IE + UE, mw1, D1, mb1,
                                                    hbuf, D1, Bn, D1, IE + UE, 1);
    // 9) x2 = relu(x1 @ mw2 + mb2)                        (K=1024, N=512)
    gemm_wmma_f32<<<grid(D2, Bn), blk, 0, stream>>>(hbuf, D1, mw2, D2, mb2,
                                                    x2buf, D2, Bn, D2, D1, 1);
    // 10) x3 = relu(x2 @ mw3 + mb3)                       (K=512, N=256)
    gemm_wmma_f32<<<grid(D3, Bn), blk, 0, stream>>>(x2buf, D2, mw3, D3, mb3,
                                                    hbuf, D3, Bn, D3, D2, 1);
    // 11) out = x3 @ mw4 + mb4                            (N=1 -> rowdot)
    rowdot_kernel<<<Bn / 8, blk, 0, stream>>>(hbuf, mw4, mb4, (float*)d_out, D3);
}